// WeightedSumSessEmbedding_60988535603950
// MI455X (gfx1250) — compile-verified
//
#include <hip/hip_runtime.h>

// WeightedSumSessEmbedding for MI455X (gfx1250):
//   out[r, :] = sum_{k in segment r} data[k] * embeddings[col_idx[k], :]
// Phase 1: rowptr[r] = lower_bound(row_idx, r) (one thread/row, latency
//          pipelined across 512 waves; row_idx is L2-resident).
// Phase 2: one wave32 per output row; async global->LDS gather of 512B
//          embedding rows, double-buffered on ASYNCcnt; scalar (SMEM) loads
//          for rowptr/col/weights via readfirstlane-uniform row id.

#define DIM 128
#define BATCH 8            // embedding rows per async batch (per wave)
#define WAVES_PER_BLOCK 8
#define BLOCK_THREADS 256  // 8 waves of 32

// b128 async copy moves one 4-DWORD vector per lane.
typedef int v4i_t __attribute__((vector_size(16)));
typedef __attribute__((address_space(1))) v4i_t* g_v4i_ptr;  // global (AS1)
typedef __attribute__((address_space(3))) v4i_t* l_v4i_ptr;  // LDS (AS3)

template <int N>
__device__ __forceinline__ void wait_async() {
  // Wait until ASYNCcnt <= N; "memory" clobber orders subsequent LDS reads.
  asm volatile("s_wait_asynccnt %0" ::"n"(N) : "memory");
}

// Copy 16B per lane (512B per wave = one embedding row) global -> LDS, async.
__device__ __forceinline__ void async_row16(const float* src, float* dst_lds) {
#if __has_builtin(__builtin_amdgcn_global_load_async_to_lds_b128)
  __builtin_amdgcn_global_load_async_to_lds_b128(
      (g_v4i_ptr)src, (l_v4i_ptr)dst_lds, /*imm offset*/ 0, /*cpol*/ 0);
#else
  unsigned lofs =
      (unsigned)(unsigned long long)(__attribute__((address_space(3))) char*)dst_lds;
  asm volatile("global_load_async_to_lds_b128 %0, %1, off" ::"v"(lofs),
               "v"(src)
               : "memory");
#endif
}

// ---------- Phase 1: rowptr[r] = lower_bound(row_idx, r), r in [0, num_rows]
__global__ void build_rowptr_kernel(const int* __restrict__ row_idx,
                                    int* __restrict__ rowptr, int nnz,
                                    int num_rows) {
  const int r = blockIdx.x * blockDim.x + threadIdx.x;
  if (r > num_rows) return;
  int lo = 0, hi = nnz;
  while (lo < hi) {
    const int mid = (lo + hi) >> 1;
    if (row_idx[mid] < r) lo = mid + 1; else hi = mid;
  }
  rowptr[r] = lo;  // rowptr[num_rows] == nnz
}

// ---------- Phase 2: weighted segment sum
__global__ __launch_bounds__(BLOCK_THREADS) void wsum_sess_emb_kernel(
    const int* __restrict__ row_idx, const int* __restrict__ col_idx,
    const float* __restrict__ data, const float* __restrict__ emb,
    float* __restrict__ out, const int* __restrict__ rowptr, int nnz,
    int num_rows) {
  // 8 waves * 2 buffers * 8 rows * 128 floats = 64KB LDS per block.
  __shared__ float smem[WAVES_PER_BLOCK * 2 * BATCH * DIM];

  const int wave = threadIdx.x >> 5;
  const int lane = threadIdx.x & 31;
  const int r = blockIdx.x * WAVES_PER_BLOCK + wave;
  if (r >= num_rows) return;  // wave-uniform exit, no barriers used

  // r is uniform within the wave; make that explicit so rowptr/col/weight
  // accesses compile to SMEM scalar loads.
  const int rs = __builtin_amdgcn_readfirstlane(r);

  int segStart, segEnd;
  if (rowptr) {
    segStart = rowptr[rs];
    segEnd = rowptr[rs + 1];
  } else {
    // Fallback (workspace too small): inline binary search.
    int lo = 0, hi = nnz;
    while (lo < hi) {
      const int mid = (lo + hi) >> 1;
      if (row_idx[mid] < rs) lo = mid + 1; else hi = mid;
    }
    segStart = lo;
    hi = nnz;
    while (lo < hi) {
      const int mid = (lo + hi) >> 1;
      if (row_idx[mid] <= rs) lo = mid + 1; else hi = mid;
    }
    segEnd = lo;
  }

  float ax = 0.f, ay = 0.f, az = 0.f, aw = 0.f;
  float* wbuf = smem + wave * (2 * BATCH * DIM);
  const int lofs = lane * 4;  // lane owns dims [4l, 4l+4)

  const int nFull = (segEnd - segStart) / BATCH;

  auto issue = [&](int b, int sel) {
    const int base = segStart + b * BATCH;
#pragma unroll
    for (int j = 0; j < BATCH; ++j) {
      const int col = col_idx[base + j];  // uniform -> scalar load
      async_row16(emb + (size_t)col * DIM + lofs,
                  wbuf + sel * (BATCH * DIM) + j * DIM + lofs);
    }
  };
  auto consume = [&](int b, int sel) {
    const int base = segStart + b * BATCH;
#pragma unroll
    for (int j = 0; j < BATCH; ++j) {
      const float w = data[base + j];  // uniform -> scalar load
      const float4 v =
          *(const float4*)(wbuf + sel * (BATCH * DIM) + j * DIM + lofs);
      ax = fmaf(w, v.x, ax);
      ay = fmaf(w, v.y, ay);
      az = fmaf(w, v.z, az);
      aw = fmaf(w, v.w, aw);
    }
  };

  if (nFull > 0) {
    issue(0, 0);
    for (int b = 0; b < nFull; ++b) {
      if (b + 1 < nFull) {
        issue(b + 1, (b + 1) & 1);
        wait_async<BATCH>();  // batch b done; batch b+1 still in flight
      } else {
        wait_async<0>();      // drain
      }
      consume(b, b & 1);
    }
  }

  // Remainder (< BATCH entries): direct global_load_b128 path.
  for (int k = segStart + nFull * BATCH; k < segEnd; ++k) {
    const float w = data[k];
    const int col = col_idx[k];
    const float4 v = *(const float4*)(emb + (size_t)col * DIM + lofs);
    ax = fmaf(w, v.x, ax);
    ay = fmaf(w, v.y, ay);
    az = fmaf(w, v.z, az);
    aw = fmaf(w, v.w, aw);
  }

  float4 res;
  res.x = ax; res.y = ay; res.z = az; res.w = aw;
  *(float4*)(out + (size_t)rs * DIM + lofs) = res;
}

extern "C" void kernel_launch(void* const* d_in, const int* in_sizes, int n_in,
                              void* d_out, int out_size, void* d_ws,
                              size_t ws_size, hipStream_t stream) {
  const int* row_idx = (const int*)d_in[0];
  const int* col_idx = (const int*)d_in[1];
  const float* data = (const float*)d_in[2];
  // d_in[3] = num_ids scalar on device; recover row count from out_size.
  const float* emb = (const float*)d_in[4];
  float* out = (float*)d_out;

  const int nnz = in_sizes[0];
  const int num_rows = out_size / DIM;

  const bool use_ws = ws_size >= (size_t)(num_rows + 1) * sizeof(int);
  int* rowptr = use_ws ? (int*)d_ws : nullptr;
  if (use_ws) {
    const int g1 = (num_rows + 1 + 255) / 256;
    build_rowptr_kernel<<<g1, 256, 0, stream>>>(row_idx, rowptr, nnz,
                                                num_rows);
  }

  const int grid = (num_rows + WAVES_PER_BLOCK - 1) / WAVES_PER_BLOCK;
  wsum_sess_emb_kernel<<<grid, BLOCK_THREADS, 0, stream>>>(
      row_idx, col_idx, data, emb, out, rowptr, nnz, num_rows);
}